// BypassP2RLoss_23656679867729
// MI455X (gfx1250) — compile-verified
//
#include <hip/hip_runtime.h>
#include <hip/hip_bf16.h>

typedef __attribute__((ext_vector_type(2))) float v2f;
typedef __attribute__((ext_vector_type(8))) float v8f;

#define HH     512
#define WW     512
#define HWSZ   (HH * WW)          // 262144
#define BATCH  64
#define NPTS   2048
#define BPB    32                 // blocks per batch for the density reduction
#define CHUNK  (HWSZ / BPB)       // 8192 floats per block
#define PB     8                  // blocks per batch for the points pass

// -------------------------------------------------------------------------
// Kernel 1: per-batch S = sum(p), Q = sum(p^2) via V_WMMA_F32_16X16X4_F32.
// A = ones(16x4) so D[m,n] = sum_k B[k,n] + C[m,n]; the full accumulator sum
// is 16x the data sum (corrected by *1/16 when writing the block partial).
// -------------------------------------------------------------------------
__global__ void __launch_bounds__(256)
p2r_reduce_kernel(const float* __restrict__ pred,
                  float* __restrict__ sp_part,   // [BATCH][BPB]
                  float* __restrict__ qp_part)   // [BATCH][BPB]
{
    const int b   = blockIdx.y;
    const int blk = blockIdx.x;
    const int tid = threadIdx.x;

    const float4* base4 =
        (const float4*)(pred + (size_t)b * HWSZ + (size_t)blk * CHUNK);

    const v2f ones = {1.0f, 1.0f};
    v8f accp = {};
    v8f accq = {};

#pragma unroll
    for (int it = 0; it < CHUNK / (256 * 4); ++it) {
        float4 d = base4[it * 256 + tid];
        v2f x0 = {d.x, d.y};
        v2f x1 = {d.z, d.w};
        v2f q0 = {d.x * d.x, d.y * d.y};
        v2f q1 = {d.z * d.z, d.w * d.w};
        accp = __builtin_amdgcn_wmma_f32_16x16x4_f32(false, ones, false, x0,
                                                     (short)0, accp, false, false);
        accp = __builtin_amdgcn_wmma_f32_16x16x4_f32(false, ones, false, x1,
                                                     (short)0, accp, false, false);
        accq = __builtin_amdgcn_wmma_f32_16x16x4_f32(false, ones, false, q0,
                                                     (short)0, accq, false, false);
        accq = __builtin_amdgcn_wmma_f32_16x16x4_f32(false, ones, false, q1,
                                                     (short)0, accq, false, false);
    }

    float lp = accp[0] + accp[1] + accp[2] + accp[3] +
               accp[4] + accp[5] + accp[6] + accp[7];
    float lq = accq[0] + accq[1] + accq[2] + accq[3] +
               accq[4] + accq[5] + accq[6] + accq[7];

    __shared__ float red[512];
    red[tid]       = lp;
    red[256 + tid] = lq;
    __syncthreads();
    for (int s = 128; s > 0; s >>= 1) {
        if (tid < s) {
            red[tid]       += red[tid + s];
            red[256 + tid] += red[256 + tid + s];
        }
        __syncthreads();
    }
    if (tid == 0) {
        sp_part[b * BPB + blk] = red[0]   * (1.0f / 16.0f);
        qp_part[b * BPB + blk] = red[256] * (1.0f / 16.0f);
    }
}

// -------------------------------------------------------------------------
// Kernel 2: per batch, X = sum_points pred[cell], G2 = sum_cells count^2
// (= # ordered point pairs sharing a cell, counted pairwise from LDS).
// -------------------------------------------------------------------------
__global__ void __launch_bounds__(256)
p2r_points_kernel(const float* __restrict__ pred,
                  const float* __restrict__ pts,
                  float* __restrict__ xp_part,   // [BATCH][PB]
                  float* __restrict__ g2_part)   // [BATCH][PB]
{
    const int b   = blockIdx.y;
    const int blk = blockIdx.x;
    const int tid = threadIdx.x;

    __shared__ int cells[NPTS];

    const float* p = pts + (size_t)b * NPTS * 2;
#pragma unroll
    for (int k = 0; k < NPTS / 256; ++k) {
        int n = k * 256 + tid;
        float px = p[2 * n];
        float py = p[2 * n + 1];
        int x = (int)floorf(px * 0.125f);
        int y = (int)floorf(py * 0.125f);
        x = x < 0 ? 0 : (x > WW - 1 ? WW - 1 : x);
        y = y < 0 ? 0 : (y > HH - 1 ? HH - 1 : y);
        cells[n] = y * WW + x;
    }
    __syncthreads();

    // This block owns points [blk*256, blk*256+256)
    const int i  = blk * 256 + tid;
    const int ci = cells[i];

    // cross term gather
    float xv = pred[(size_t)b * HWSZ + ci];

    // pairwise same-cell count (LDS broadcast reads, int4 vectorized)
    int cnt = 0;
    const int4* c4 = (const int4*)cells;
#pragma unroll 4
    for (int j4 = 0; j4 < NPTS / 4; ++j4) {
        int4 c = c4[j4];
        cnt += (c.x == ci) + (c.y == ci) + (c.z == ci) + (c.w == ci);
    }

    __shared__ float rx[256];
    __shared__ int   rc[256];
    rx[tid] = xv;
    rc[tid] = cnt;
    __syncthreads();
    for (int s = 128; s > 0; s >>= 1) {
        if (tid < s) {
            rx[tid] += rx[tid + s];
            rc[tid] += rc[tid + s];
        }
        __syncthreads();
    }
    if (tid == 0) {
        xp_part[b * PB + blk] = rx[0];
        g2_part[b * PB + blk] = (float)rc[0];
    }
}

// -------------------------------------------------------------------------
// Kernel 3: fold partials, compute the 6 outputs.
// -------------------------------------------------------------------------
__global__ void __launch_bounds__(64)
p2r_finalize_kernel(const float* __restrict__ sp, const float* __restrict__ qp,
                    const float* __restrict__ xp, const float* __restrict__ g2p,
                    float* __restrict__ out)
{
    const int b = threadIdx.x;   // 0..63 == batch index

    float S = 0.0f, Q = 0.0f;
    for (int k = 0; k < BPB; ++k) { S += sp[b * BPB + k]; Q += qp[b * BPB + k]; }
    float X = 0.0f, G2 = 0.0f;
    for (int k = 0; k < PB; ++k) { X += xp[b * PB + k]; G2 += g2p[b * PB + k]; }

    const float gt         = (float)NPTS;                 // 2048
    float pred_count = S * (1.0f / 64.0f);                // /cell_area
    float err        = fabsf(pred_count - gt);
    float Sden       = S + 1e-8f;
    float Tden       = gt + 1e-8f;
    float spatial    = (Q / (Sden * Sden)
                        - 2.0f * X / (Sden * Tden)
                        + G2 / (Tden * Tden)) * (1.0f / (float)HWSZ);

    __shared__ float re[64], rs[64], rp[64];
    re[b] = err; rs[b] = spatial; rp[b] = pred_count;
    __syncthreads();
    for (int s = 32; s > 0; s >>= 1) {
        if (b < s) { re[b] += re[b + s]; rs[b] += rs[b + s]; rp[b] += rp[b + s]; }
        __syncthreads();
    }
    if (b == 0) {
        float count_loss   = re[0] * (1.0f / (float)BATCH);
        float scale_loss   = count_loss / (gt + 1e-6f);
        float spatial_loss = rs[0] * (1.0f / (float)BATCH);
        float mae          = count_loss;
        float bias         = rp[0] / (gt * (float)BATCH);
        float total        = 2.0f * count_loss + 0.5f * scale_loss
                             + 0.15f * spatial_loss;
        out[0] = total;
        out[1] = count_loss;
        out[2] = scale_loss;
        out[3] = spatial_loss;
        out[4] = mae;
        out[5] = bias;
    }
}

extern "C" void kernel_launch(void* const* d_in, const int* in_sizes, int n_in,
                              void* d_out, int out_size, void* d_ws, size_t ws_size,
                              hipStream_t stream) {
    const float* pred = (const float*)d_in[0];   // [B,1,H,W] f32
    const float* pts  = (const float*)d_in[1];   // [B,N,2]  f32
    // d_in[2] = cell_area (64) — compile-time constant here.

    float* ws = (float*)d_ws;                    // 20 KB of partials, no init needed
    float* sp = ws;                              // [64][32]
    float* qp = sp + BATCH * BPB;                // [64][32]
    float* xp = qp + BATCH * BPB;                // [64][8]
    float* g2 = xp + BATCH * PB;                 // [64][8]

    p2r_reduce_kernel<<<dim3(BPB, BATCH), 256, 0, stream>>>(pred, sp, qp);
    p2r_points_kernel<<<dim3(PB, BATCH), 256, 0, stream>>>(pred, pts, xp, g2);
    p2r_finalize_kernel<<<1, 64, 0, stream>>>(sp, qp, xp, g2, (float*)d_out);
}